// LGMLoss_12893491823014
// MI455X (gfx1250) — compile-verified
//
#include <hip/hip_runtime.h>
#include <math.h>

#define B_N 8192
#define C_N 1000
#define D_N 128

__constant__ const float kALPHA  = 0.5f;
__constant__ const float kLAMBDA = 0.5f;
__constant__ const float kLAM2   = 0.1f;
__constant__ const float kTINV   = 1.0f / 0.3f;

typedef float v2f __attribute__((ext_vector_type(2)));
typedef float v8f __attribute__((ext_vector_type(8)));

// ---------------------------------------------------------------------------
// K=128 f32 WMMA panel dot: two independent accumulator chains (even/odd k)
// so the matrix pipe is not serialized on a single C->D RAW chain.
// a[k] holds A[m][4k + 2*half + {0,1}]; brow points at B column row start
// (+2*half), per-lane.  Unmasked fast path: all 16 columns valid.
// ---------------------------------------------------------------------------
__device__ __forceinline__ v8f wmma_k128(const v2f* __restrict__ a,
                                         const float* __restrict__ brow) {
    v8f acc0 = {}, acc1 = {};
#pragma unroll
    for (int k = 0; k < 32; k += 2) {
        const v2f b0 = *(const v2f*)(brow + 4 * k);
        const v2f b1 = *(const v2f*)(brow + 4 * (k + 1));
        acc0 = __builtin_amdgcn_wmma_f32_16x16x4_f32(
            false, a[k],     false, b0, (short)0, acc0, false, false);
        acc1 = __builtin_amdgcn_wmma_f32_16x16x4_f32(
            false, a[k + 1], false, b1, (short)0, acc1, false, false);
    }
    return acc0 + acc1;
}

// Masked variant for the single ragged tail tile (lane-level zero fill; the
// WMMA itself still executes with EXEC all ones).
__device__ __forceinline__ v8f wmma_k128_masked(const v2f* __restrict__ a,
                                                const float* __restrict__ brow,
                                                bool colok) {
    v8f acc0 = {}, acc1 = {};
#pragma unroll
    for (int k = 0; k < 32; k += 2) {
        v2f b0 = *(const v2f*)(brow + 4 * k);
        v2f b1 = *(const v2f*)(brow + 4 * (k + 1));
        if (!colok) { b0.x = 0.f; b0.y = 0.f; b1.x = 0.f; b1.y = 0.f; }
        acc0 = __builtin_amdgcn_wmma_f32_16x16x4_f32(
            false, a[k],     false, b0, (short)0, acc0, false, false);
        acc1 = __builtin_amdgcn_wmma_f32_16x16x4_f32(
            false, a[k + 1], false, b1, (short)0, acc1, false, false);
    }
    return acc0 + acc1;
}

// ---------------------------------------------------------------------------
// Zero the scalar accumulators (ce_sum, margin_sum, scl_sum, n_valid)
// ---------------------------------------------------------------------------
__global__ void zero_accum_kernel(float* acc) {
    if (threadIdx.x < 4) acc[threadIdx.x] = 0.0f;
}

// ---------------------------------------------------------------------------
// Row L2 normalize. raw_ss=1 -> store raw sum of squares (YY for means);
// raw_ss=0 -> store sum of squares of the normalized row (XX for feat).
// grid = nrows, block = 128 (== D_N)
// ---------------------------------------------------------------------------
__global__ void norm_rows_kernel(const float* __restrict__ x,
                                 float* __restrict__ xn,
                                 float* __restrict__ ss, int raw_ss) {
    __shared__ float red[128];
    const int row = blockIdx.x;
    const int t   = threadIdx.x;
    const float v = x[(size_t)row * D_N + t];
    red[t] = v * v;
    __syncthreads();
    for (int s = 64; s > 0; s >>= 1) {
        if (t < s) red[t] += red[t + s];
        __syncthreads();
    }
    const float sq = red[0];
    const float n  = fmaxf(sqrtf(sq), 1e-12f);
    xn[(size_t)row * D_N + t] = v / n;
    if (t == 0 && ss) ss[row] = raw_ss ? sq : sq / (n * n);
}

// ---------------------------------------------------------------------------
// neg_sqr_dist = -0.5 * (XX - 2*feat_n@means^T + YY)  via f32 WMMA 16x16x4.
// grid = B/16 row tiles, block = 256 (8 waves); wave w handles col tiles
// w, w+8, ...  A panel (16x128) kept in 64 VGPRs per lane across the loop.
// Full tiles (ct < 62) run unmasked; the ragged tail tile runs masked.
// ---------------------------------------------------------------------------
__global__ void nsd_gemm_kernel(const float* __restrict__ feat_n,
                                const float* __restrict__ means_n,
                                const float* __restrict__ XX,
                                const float* __restrict__ YY,
                                float* __restrict__ out) {
    const int lane = threadIdx.x & 31;
    const int wave = threadIdx.x >> 5;
    const int i0   = blockIdx.x * 16;
    const int m    = lane & 15;   // A row / D col within tile
    const int half = lane >> 4;   // K pair selector / D row-half selector

    constexpr int CT_FULL = C_N / 16;   // 62 full column tiles

    // Preload A: a[k] = feat_n[i0+m][4k + 2*half + {0,1}]
    v2f a[32];
    const float* arow = feat_n + (size_t)(i0 + m) * D_N + 2 * half;
#pragma unroll
    for (int k = 0; k < 32; ++k) a[k] = *(const v2f*)(arow + 4 * k);

    // Loop-invariant: XX for the 8 D-rows this lane holds (per vgpr slot r).
    float xxv[8];
#pragma unroll
    for (int r = 0; r < 8; ++r) xxv[r] = XX[i0 + 8 * half + r];

    for (int ct = wave; ct < CT_FULL; ct += 8) {
        const int col = ct * 16 + m;
        const float* brow = means_n + (size_t)col * D_N + 2 * half;
        const v8f d = wmma_k128(a, brow);

        const float yy = YY[col];
#pragma unroll
        for (int r = 0; r < 8; ++r) {
            const int row = i0 + 8 * half + r;   // D row held in vgpr r
            out[(size_t)row * C_N + col] = -0.5f * (xxv[r] - 2.0f * d[r] + yy);
        }
    }

    // Ragged tail tile (cols 992..999 valid) — one wave per block.
    for (int ct = CT_FULL; ct * 16 < C_N; ++ct) {
        if ((ct & 7) != wave) continue;
        const int  col   = ct * 16 + m;
        const bool colok = (col < C_N);
        const int  colc  = colok ? col : (C_N - 1);
        const float* brow = means_n + (size_t)colc * D_N + 2 * half;
        const v8f d = wmma_k128_masked(a, brow, colok);
        if (colok) {
            const float yy = YY[col];
#pragma unroll
            for (int r = 0; r < 8; ++r) {
                const int row = i0 + 8 * half + r;
                out[(size_t)row * C_N + col] =
                    -0.5f * (xxv[r] - 2.0f * d[r] + yy);
            }
        }
    }
}

// ---------------------------------------------------------------------------
// Per-row cross entropy with margin factor (1+alpha) on the true class.
// grid = B, block = 256.
// ---------------------------------------------------------------------------
__global__ void ce_kernel(const float* __restrict__ nsd,
                          const int* __restrict__ labels,
                          float* __restrict__ ce_sum) {
    __shared__ float red[256];
    const int b = blockIdx.x;
    const int t = threadIdx.x;
    const float* row = nsd + (size_t)b * C_N;
    const int lab = labels[b];

    float mx = -3.4e38f;
    for (int c = t; c < C_N; c += 256) {
        const float l = row[c] * ((c == lab) ? (1.0f + kALPHA) : 1.0f);
        mx = fmaxf(mx, l);
    }
    red[t] = mx; __syncthreads();
    for (int s = 128; s > 0; s >>= 1) {
        if (t < s) red[t] = fmaxf(red[t], red[t + s]);
        __syncthreads();
    }
    mx = red[0]; __syncthreads();

    float se = 0.0f;
    for (int c = t; c < C_N; c += 256) {
        const float l = row[c] * ((c == lab) ? (1.0f + kALPHA) : 1.0f);
        se += __expf(l - mx);
    }
    red[t] = se; __syncthreads();
    for (int s = 128; s > 0; s >>= 1) {
        if (t < s) red[t] += red[t + s];
        __syncthreads();
    }
    if (t == 0) {
        const float ll   = row[lab] * (1.0f + kALPHA);
        const float logp = ll - mx - __logf(red[0]);
        atomicAdd(ce_sum, -logp);
    }
}

// ---------------------------------------------------------------------------
// Margin term: sum_b ||feat_n[b] - means_n[label[b]]||^2, grid=B, block=128.
// ---------------------------------------------------------------------------
__global__ void margin_kernel(const float* __restrict__ feat_n,
                              const float* __restrict__ means_n,
                              const int* __restrict__ labels,
                              float* __restrict__ margin_sum) {
    __shared__ float red[128];
    const int b = blockIdx.x;
    const int t = threadIdx.x;
    const float d = feat_n[(size_t)b * D_N + t] -
                    means_n[(size_t)labels[b] * D_N + t];
    red[t] = d * d;
    __syncthreads();
    for (int s = 64; s > 0; s >>= 1) {
        if (t < s) red[t] += red[t + s];
        __syncthreads();
    }
    if (t == 0) atomicAdd(margin_sum, red[0]);
}

// ---------------------------------------------------------------------------
// SCL loss, flash-style streaming over the BxB sim matrix (never stored).
// grid = B/16 row tiles, block = 256 (8 waves). Each wave streams column
// tiles w, w+8, ... computing sim via f32 WMMA, accumulating per-row:
//   bottom = sum_{k!=i} exp(sim_ik), cnt = #pos, psim = sum_pos sim.
// Reduced through LDS float atomics; per_i = (psim - cnt*log(bottom))/cnt.
// ---------------------------------------------------------------------------
__global__ void scl_kernel(const float* __restrict__ feat_n,
                           const int* __restrict__ labels,
                           float* __restrict__ scl_sum,
                           float* __restrict__ n_valid) {
    __shared__ float s_bot[16];
    __shared__ float s_cnt[16];
    __shared__ float s_psim[16];

    const int lane = threadIdx.x & 31;
    const int wave = threadIdx.x >> 5;
    const int i0   = blockIdx.x * 16;
    const int m    = lane & 15;
    const int half = lane >> 4;

    if (threadIdx.x < 16) {
        s_bot[threadIdx.x]  = 0.0f;
        s_cnt[threadIdx.x]  = 0.0f;
        s_psim[threadIdx.x] = 0.0f;
    }
    __syncthreads();

    // Preload A panel (16 rows x 128) and the 8 row labels this lane covers.
    v2f a[32];
    const float* arow = feat_n + (size_t)(i0 + m) * D_N + 2 * half;
#pragma unroll
    for (int k = 0; k < 32; ++k) a[k] = *(const v2f*)(arow + 4 * k);

    int rlab[8];
#pragma unroll
    for (int r = 0; r < 8; ++r) rlab[r] = labels[i0 + 8 * half + r];

    float lbot[8], lcnt[8], lpsim[8];
#pragma unroll
    for (int r = 0; r < 8; ++r) { lbot[r] = 0.0f; lcnt[r] = 0.0f; lpsim[r] = 0.0f; }

    for (int ct = wave; ct < B_N / 16; ct += 8) {
        const int col = ct * 16 + m;
        const float* brow = feat_n + (size_t)col * D_N + 2 * half;

        // Warm L2/L0 for the next column tile this wave will touch
        // (gfx1250 global_prefetch path; speculative, safely droppable).
        if (ct + 8 < B_N / 16) {
            __builtin_prefetch(brow + 8 * 16 * D_N, 0, 1);
        }

        const v8f d = wmma_k128(a, brow);

        const int clab = labels[col];
#pragma unroll
        for (int r = 0; r < 8; ++r) {
            const int   row  = i0 + 8 * half + r;
            const float sim  = d[r] * kTINV;
            const bool  diag = (row == col);
            lbot[r] += diag ? 0.0f : __expf(sim);
            if (!diag && rlab[r] == clab) {
                lcnt[r]  += 1.0f;
                lpsim[r] += sim;
            }
        }
    }

#pragma unroll
    for (int r = 0; r < 8; ++r) {
        const int mr = 8 * half + r;
        atomicAdd(&s_bot[mr],  lbot[r]);
        atomicAdd(&s_cnt[mr],  lcnt[r]);
        atomicAdd(&s_psim[mr], lpsim[r]);
    }
    __syncthreads();

    if (threadIdx.x < 16) {
        const float cnt = s_cnt[threadIdx.x];
        if (cnt > 0.5f) {
            const float per = (s_psim[threadIdx.x] -
                               cnt * __logf(s_bot[threadIdx.x])) / cnt;
            atomicAdd(scl_sum, per);
            atomicAdd(n_valid, 1.0f);
        }
    }
}

// ---------------------------------------------------------------------------
// loss = (1-lam2)*ce + lam2*scl + lambda*margin
// ---------------------------------------------------------------------------
__global__ void finalize_kernel(const float* __restrict__ acc,
                                float* __restrict__ loss_out) {
    const float ce     = acc[0] / (float)B_N;
    const float margin = acc[1] / (2.0f * (float)B_N);
    const float nv     = fmaxf(acc[3], 1.0f);
    const float scl    = -acc[2] / nv;
    loss_out[0] = (1.0f - kLAM2) * ce + kLAM2 * scl + kLAMBDA * margin;
}

// ---------------------------------------------------------------------------
extern "C" void kernel_launch(void* const* d_in, const int* in_sizes, int n_in,
                              void* d_out, int out_size, void* d_ws, size_t ws_size,
                              hipStream_t stream) {
    const float* feat   = (const float*)d_in[0];
    const int*   labels = (const int*)d_in[1];
    const float* means  = (const float*)d_in[2];
    float*       out    = (float*)d_out;   // [B*C] nsd, then [1] loss

    float* feat_n  = (float*)d_ws;                       // B*D
    float* means_n = feat_n + (size_t)B_N * D_N;         // C*D
    float* XX      = means_n + (size_t)C_N * D_N;        // B
    float* YY      = XX + B_N;                           // C
    float* acc     = YY + C_N;                           // 4 scalars

    zero_accum_kernel<<<1, 32, 0, stream>>>(acc);
    norm_rows_kernel<<<B_N, 128, 0, stream>>>(feat, feat_n, XX, /*raw=*/0);
    norm_rows_kernel<<<C_N, 128, 0, stream>>>(means, means_n, YY, /*raw=*/1);
    nsd_gemm_kernel<<<B_N / 16, 256, 0, stream>>>(feat_n, means_n, XX, YY, out);
    ce_kernel<<<B_N, 256, 0, stream>>>(out, labels, acc + 0);
    margin_kernel<<<B_N, 128, 0, stream>>>(feat_n, means_n, labels, acc + 1);
    scl_kernel<<<B_N / 16, 256, 0, stream>>>(feat_n, labels, acc + 2, acc + 3);
    finalize_kernel<<<1, 1, 0, stream>>>(acc, out + (size_t)B_N * C_N);
}